// Sparsemax_28192165330961
// MI455X (gfx1250) — compile-verified
//
#include <hip/hip_runtime.h>
#include <hip/hip_bf16.h>
#include <cstdint>

#define DCH   1024
#define LLEN  1024
#define BATCH 32
#define KW    7
#define HDIM  32
#define OUTC  11
#define NEGV  (-1e9f)
#define BNEPS 1e-5f

#define TM 128        // d_out rows per block (8 waves x 16)
#define TN 64         // l columns per block (4 N-tiles per wave)
#define NTILE (TN / 16)
#define DI 32         // d_in block staged in LDS
#define XS_COLS (TN + KW - 1)   // 70 halo columns
#define XS_STRIDE 34  // padded d_in stride: even (b64-aligned), 17 coprime 32 (bank-conflict-free)
#define NGROUP (KW * (DI / 4))  // 56 wmma groups per di0 block

typedef __attribute__((ext_vector_type(2))) float v2f;
typedef __attribute__((ext_vector_type(8))) float v8f;

#if defined(__has_builtin)
#if __has_builtin(__builtin_amdgcn_global_load_async_to_lds_b32)
#define ASYNC_COPY 1
#endif
#endif

typedef __attribute__((address_space(1))) int g_int;
typedef __attribute__((address_space(3))) int l_int;

// ---------------------------------------------------------------------------
// Kernel 0: transpose conv weights W[o][i][k] -> Wt[k][i][o] (both tensors)
// ---------------------------------------------------------------------------
__global__ __launch_bounds__(256)
void transpose_w_kernel(const float* __restrict__ w_conv,
                        const float* __restrict__ w_att,
                        float* __restrict__ wt_conv,
                        float* __restrict__ wt_att) {
  const float* src = (blockIdx.y == 0) ? w_conv : w_att;
  float*       dst = (blockIdx.y == 0) ? wt_conv : wt_att;
  unsigned idx  = blockIdx.x * 256u + threadIdx.x;  // 0 .. 7*2^20-1
  unsigned k    = idx >> 20;
  unsigned rem  = idx & 0xFFFFFu;
  unsigned din  = rem >> 10;
  unsigned dout = rem & 1023u;
  dst[idx] = src[(size_t)dout * (DCH * KW) + din * KW + k];
}

// ---------------------------------------------------------------------------
// Kernel 1: fused dual conv1d via V_WMMA_F32_16X16X4_F32.
// Block = 256 threads (8 wave32). Tile: TM x TN of both o and att for one b.
// x tile double-buffered in LDS via GLOBAL_LOAD_ASYNC_TO_LDS (ASYNCcnt).
// A fragments software-pipelined from L2-resident transposed weights.
// ---------------------------------------------------------------------------
__global__ __launch_bounds__(256)
void conv_wmma_kernel(const float* __restrict__ x,
                      const unsigned char* __restrict__ mask,
                      const float* __restrict__ wtO,   // [KW][DCH][DCH]
                      const float* __restrict__ wtA,   // [KW][DCH][DCH]
                      const float* __restrict__ biasO,
                      const float* __restrict__ biasA,
                      float* __restrict__ o_out,
                      float* __restrict__ att_out) {
  __shared__ __align__(16) float xs[2][XS_COLS * XS_STRIDE];

  const int b     = blockIdx.x;
  const int l0    = blockIdx.y * TN;
  const int dout0 = blockIdx.z * TM;
  const int tid   = threadIdx.x;
  const int wave  = tid >> 5;
  const int lane  = tid & 31;
  const int m     = lane & 15;   // M (A) / N (B,C,D) index within half-wave
  const int hi    = lane >> 4;   // half-wave: selects K-pair / output row block

  const int doutRow = dout0 + wave * 16 + m;     // A-fragment column (coalesced)
  const float* xb = x + (size_t)b * DCH * LLEN;

  // Zero-fill out-of-range halo slots once (both buffers); async loads only
  // ever touch in-range slots, so zeros persist across all di0 iterations.
  for (int e = tid; e < XS_COLS * DI; e += 256) {
    int din = e / XS_COLS;
    int col = e - din * XS_COLS;
    int l   = l0 + col - (KW / 2);
    if (l < 0 || l >= LLEN) {
      xs[0][col * XS_STRIDE + din] = 0.0f;
      xs[1][col * XS_STRIDE + din] = 0.0f;
    }
  }

  // Stage x tile for d_in block di0n into buffer nb.
  auto stage = [&](int di0n, int nb) {
    for (int e = tid; e < XS_COLS * DI; e += 256) {
      int din = e / XS_COLS;
      int col = e - din * XS_COLS;       // consecutive tid -> consecutive l
      int l   = l0 + col - (KW / 2);
      if (l >= 0 && l < LLEN) {
        const float* gp = xb + (size_t)(di0n + din) * LLEN + l;
        float*       lp = &xs[nb][col * XS_STRIDE + din];
#ifdef ASYNC_COPY
        __builtin_amdgcn_global_load_async_to_lds_b32(
            (g_int*)gp, (l_int*)lp, 0, 0);
#else
        *lp = *gp;
#endif
      }
    }
  };

  stage(0, 0);   // prologue: first tile in flight

  v8f accO[NTILE] = {};
  v8f accA[NTILE] = {};

#define A_IDX(g) (((size_t)((((g) >> 3) * DCH) + di0 + (((g) & 7) << 2) + 2 * hi)) * DCH + doutRow)

  for (int di0 = 0; di0 < DCH; di0 += DI) {
    const int cur = (di0 / DI) & 1;
#ifdef ASYNC_COPY
    asm volatile("s_wait_asynccnt 0" ::: "memory");
#endif
    __syncthreads();                       // cur buffer staged & visible
    if (di0 + DI < DCH) stage(di0 + DI, cur ^ 1);   // overlap next tile with compute

    // Software-pipelined A fragments over NGROUP (k, d_in-quad) groups.
    size_t ia0 = A_IDX(0);
    v2f aO_n, aA_n;
    aO_n.x = wtO[ia0]; aO_n.y = wtO[ia0 + DCH];
    aA_n.x = wtA[ia0]; aA_n.y = wtA[ia0 + DCH];

#pragma unroll
    for (int g = 0; g < NGROUP; ++g) {
      v2f aO = aO_n, aA = aA_n;
      if (g + 1 < NGROUP) {
        size_t ia = A_IDX(g + 1);
        aO_n.x = wtO[ia]; aO_n.y = wtO[ia + DCH];
        aA_n.x = wtA[ia]; aA_n.y = wtA[ia + DCH];
      }
      const int kk = g >> 3;
      const int dd = ((g & 7) << 2) + 2 * hi;
      const float* bcol = &xs[cur][(m + kk) * XS_STRIDE + dd];
#pragma unroll
      for (int t = 0; t < NTILE; ++t) {
        v2f bv = *(const v2f*)(bcol + t * 16 * XS_STRIDE);   // ds_load_b64
        accO[t] = __builtin_amdgcn_wmma_f32_16x16x4_f32(false, aO, false, bv,
                                                        (short)0, accO[t], false, false);
        accA[t] = __builtin_amdgcn_wmma_f32_16x16x4_f32(false, aA, false, bv,
                                                        (short)0, accA[t], false, false);
      }
    }
  }
#undef A_IDX

  // Epilogue. C/D layout: reg r holds row (r + 8*hi), col = m + 16*t.
  bool mk[NTILE];
#pragma unroll
  for (int t = 0; t < NTILE; ++t)
    mk[t] = mask[(size_t)b * LLEN + l0 + m + 16 * t] != 0;

  const size_t obase = (size_t)b * DCH * LLEN;
#pragma unroll
  for (int r = 0; r < 8; ++r) {
    const int dout = dout0 + wave * 16 + r + 8 * hi;
    const float bo = biasO[dout];
    const float ba = biasA[dout];
    const size_t row = obase + (size_t)dout * LLEN + l0 + m;
#pragma unroll
    for (int t = 0; t < NTILE; ++t) {
      o_out[row + 16 * t]   = accO[t][r] + bo;
      att_out[row + 16 * t] = mk[t] ? (accA[t][r] + ba) : NEGV;
    }
  }
}

// ---------------------------------------------------------------------------
// Block-wide reductions: wave32 shfl_xor + 8-entry LDS combine (2 barriers).
// All threads receive the result.
// ---------------------------------------------------------------------------
__device__ __forceinline__ float block_sum(float v, float* red8, int tid) {
#pragma unroll
  for (int off = 16; off > 0; off >>= 1) v += __shfl_xor(v, off, 32);
  if ((tid & 31) == 0) red8[tid >> 5] = v;
  __syncthreads();
  float t = red8[0] + red8[1] + red8[2] + red8[3] +
            red8[4] + red8[5] + red8[6] + red8[7];
  __syncthreads();
  return t;
}

__device__ __forceinline__ float block_max(float v, float* red8, int tid) {
#pragma unroll
  for (int off = 16; off > 0; off >>= 1) v = fmaxf(v, __shfl_xor(v, off, 32));
  if ((tid & 31) == 0) red8[tid >> 5] = v;
  __syncthreads();
  float t = fmaxf(fmaxf(fmaxf(red8[0], red8[1]), fmaxf(red8[2], red8[3])),
                  fmaxf(fmaxf(red8[4], red8[5]), fmaxf(red8[6], red8[7])));
  __syncthreads();
  return t;
}

// ---------------------------------------------------------------------------
// Kernel 2: per-(b,d) sparsemax over L via bisection on sum(relu(z - tau)) = 1,
// then o1 = sum(o*p), o2 = max(o). h[b, d] = o1, h[b, D + d] = o2.
// ---------------------------------------------------------------------------
__global__ __launch_bounds__(256)
void sparsemax_kernel(const float* __restrict__ o_in,
                      const float* __restrict__ att_in,
                      float* __restrict__ h) {
  __shared__ float zs[LLEN];
  __shared__ float os[LLEN];
  __shared__ float red8[8];

  const int d = blockIdx.x;
  const int b = blockIdx.y;
  const int tid = threadIdx.x;
  const size_t base = ((size_t)b * DCH + d) * LLEN;

  float mloc = -3.4e38f;
  for (int i = tid; i < LLEN; i += 256) {
    float z = att_in[base + i];
    zs[i] = z;
    os[i] = o_in[base + i];
    mloc = fmaxf(mloc, z);
  }
  __syncthreads();                 // zs/os staged before anyone reduces
  const float zmax = block_max(mloc, red8, tid);

  // f(tau) = sum relu(z - tau): piecewise-linear, decreasing;
  // f(zmax)=0 < 1 <= f(zmax-1). 32 bisections -> interval ~2^-32.
  float lo = zmax - 1.0f, hi = zmax;
  for (int it = 0; it < 32; ++it) {
    const float mid = 0.5f * (lo + hi);
    float s = 0.0f;
    for (int i = tid; i < LLEN; i += 256) s += fmaxf(zs[i] - mid, 0.0f);
    const float tot = block_sum(s, red8, tid);
    if (tot >= 1.0f) lo = mid; else hi = mid;   // uniform across block
  }
  const float tau = 0.5f * (lo + hi);

  float s1 = 0.0f, m2 = -3.4e38f;
  for (int i = tid; i < LLEN; i += 256) {
    const float p  = fmaxf(zs[i] - tau, 0.0f);
    const float ov = os[i];
    s1 += p * ov;
    m2 = fmaxf(m2, ov);
  }
  const float o1 = block_sum(s1, red8, tid);
  const float o2 = block_max(m2, red8, tid);
  if (tid == 0) {
    h[(size_t)b * (2 * DCH) + d]       = o1;
    h[(size_t)b * (2 * DCH) + DCH + d] = o2;
  }
}

// ---------------------------------------------------------------------------
// Kernel 3: head: h[32,2048] @ lin_w.T + lin_b -> relu -> BN -> @ out_w.T + out_b
// Single block, 1024 threads (32 rows x 32 hidden units).
// ---------------------------------------------------------------------------
__global__ __launch_bounds__(1024)
void head_kernel(const float* __restrict__ h,
                 const float* __restrict__ lin_w, const float* __restrict__ lin_b,
                 const float* __restrict__ gamma, const float* __restrict__ beta,
                 const float* __restrict__ mean,  const float* __restrict__ var,
                 const float* __restrict__ out_w, const float* __restrict__ out_b,
                 float* __restrict__ out) {
  __shared__ float hm[BATCH][HDIM];
  const int tid = threadIdx.x;
  const int row = tid >> 5;
  const int j   = tid & 31;

  const float* hr = h + (size_t)row * (2 * DCH);
  const float* wr = lin_w + (size_t)j * (2 * DCH);
  float acc = lin_b[j];
  for (int c = 0; c < 2 * DCH; ++c) acc += hr[c] * wr[c];
  acc = fmaxf(acc, 0.0f);
  acc = (acc - mean[j]) * rsqrtf(var[j] + BNEPS) * gamma[j] + beta[j];
  hm[row][j] = acc;
  __syncthreads();

  if (tid < BATCH * OUTC) {
    const int r  = tid / OUTC;
    const int oi = tid - r * OUTC;
    float a = out_b[oi];
    for (int c = 0; c < HDIM; ++c) a += hm[r][c] * out_w[oi * HDIM + c];
    out[r * OUTC + oi] = a;
  }
}

// ---------------------------------------------------------------------------
extern "C" void kernel_launch(void* const* d_in, const int* in_sizes, int n_in,
                              void* d_out, int out_size, void* d_ws, size_t ws_size,
                              hipStream_t stream) {
  const float*         x        = (const float*)d_in[0];
  const unsigned char* mask     = (const unsigned char*)d_in[1];
  const float*         conv1_w  = (const float*)d_in[2];
  const float*         conv1_b  = (const float*)d_in[3];
  const float*         attend_w = (const float*)d_in[4];
  const float*         attend_b = (const float*)d_in[5];
  const float*         lin_w    = (const float*)d_in[6];
  const float*         lin_b    = (const float*)d_in[7];
  const float*         bn_gamma = (const float*)d_in[8];
  const float*         bn_beta  = (const float*)d_in[9];
  const float*         bn_mean  = (const float*)d_in[10];
  const float*         bn_var   = (const float*)d_in[11];
  const float*         out_w    = (const float*)d_in[12];
  const float*         out_b    = (const float*)d_in[13];

  // Workspace (floats): wtO | wtA | obuf | attbuf | hbuf  (~312 MiB total)
  const size_t WT   = (size_t)KW * DCH * DCH;     // 7,340,032
  const size_t OBUF = (size_t)BATCH * DCH * LLEN; // 33,554,432
  float* ws     = (float*)d_ws;
  float* wtO    = ws;
  float* wtA    = wtO + WT;
  float* obuf   = wtA + WT;
  float* attbuf = obuf + OBUF;
  float* hbuf   = attbuf + OBUF;

  transpose_w_kernel<<<dim3((unsigned)(WT / 256), 2), 256, 0, stream>>>(
      conv1_w, attend_w, wtO, wtA);

  conv_wmma_kernel<<<dim3(BATCH, LLEN / TN, DCH / TM), 256, 0, stream>>>(
      x, mask, wtO, wtA, conv1_b, attend_b, obuf, attbuf);

  sparsemax_kernel<<<dim3(DCH, BATCH), 256, 0, stream>>>(obuf, attbuf, hbuf);

  head_kernel<<<1, 1024, 0, stream>>>(hbuf, lin_w, lin_b, bn_gamma, bn_beta,
                                      bn_mean, bn_var, out_w, out_b,
                                      (float*)d_out);
}